// PointLaplacianLoss_1382979470103
// MI455X (gfx1250) — compile-verified
//
#include <hip/hip_runtime.h>
#include <hip/hip_bf16.h>

typedef __attribute__((ext_vector_type(2))) float v2f;
typedef __attribute__((ext_vector_type(8))) float v8f;

#define BATCH 4
#define NPTS 8192
#define KNN 8
#define WAVES_PER_BLOCK 4
#define LDS_STRIDE 17 // 16 cols + 1 pad float -> all 32 lanes hit distinct banks

// ---------------------------------------------------------------------------
// Kernel 1: squared norms of point1 rows
// ---------------------------------------------------------------------------
__global__ void PLL_sqn_kernel(const float* __restrict__ p, float* __restrict__ sqn) {
    int t = blockIdx.x * blockDim.x + threadIdx.x;
    if (t < BATCH * NPTS) {
        float x = p[t * 3 + 0], y = p[t * 3 + 1], z = p[t * 3 + 2];
        sqn[t] = x * x + y * y + z * z;
    }
}

// ---------------------------------------------------------------------------
// Kernel 2: fused distance (WMMA Gram matrix) + per-row top-8 selection.
// One wave owns 32 rows of one batch; streams 16-column tiles of the distance
// matrix via V_WMMA_F32_16X16X4_F32 (K=3 padded to 4), transposes through LDS,
// each lane keeps a sorted top-8 (self excluded via diagonal poisoning) in
// registers. Hot loop: branchless global loads, batched LDS row read (one
// dscnt wait), pure-VALU guarded insertion.
// ---------------------------------------------------------------------------
__global__ __launch_bounds__(WAVES_PER_BLOCK * 32)
void PLL_knn_kernel(const float* __restrict__ pts, const float* __restrict__ sqn,
                    int* __restrict__ knn) {
    __shared__ float lds[WAVES_PER_BLOCK * 32 * LDS_STRIDE];

    const int   lane  = threadIdx.x & 31;
    const int   wave  = threadIdx.x >> 5;
    const int   g     = blockIdx.x * WAVES_PER_BLOCK + wave; // row-group id
    const int   b     = g >> 8;            // NPTS/32 = 256 row-groups per batch
    const int   row0  = (g & 255) << 5;    // 32 rows per wave
    const bool  hi    = lane >= 16;
    const int   l16   = lane & 15;
    const int   kofs  = hi ? 2 : 0;        // lane's first K component: x or z
    const float ymask = hi ? 0.0f : 1.0f;  // zero pad for K=1/3 slot

    const float* P  = pts + (size_t)b * NPTS * 3;
    const float* SQ = sqn + (size_t)b * NPTS;

    // A operands: 16x4 f32 layout — lanes 0-15 hold (K=0,K=1)=(x,y) of row
    // M=lane, lanes 16-31 hold (K=2,K=3)=(z,0) of row M=lane-16.
    v2f a_lo, a_hi;
    {
        int r3 = (row0 + l16) * 3;
        a_lo[0] = P[r3 + kofs];
        a_lo[1] = P[r3 + 1] * ymask;
        a_hi[0] = P[r3 + 48 + kofs];          // +16 rows = +48 floats
        a_hi[1] = P[r3 + 48 + 1] * ymask;
    }

    const int   myrow = row0 + lane;
    const float sqr   = SQ[myrow];

    float bd[KNN];
    int   bi[KNN];
#pragma unroll
    for (int j = 0; j < KNN; ++j) { bd[j] = 3.4e38f; bi[j] = 0; }

    float* T          = lds + wave * 32 * LDS_STRIDE;
    float* Trow       = T + lane * LDS_STRIDE; // lane's scan row
    const int diagCol = myrow & 15;
    const int diagT   = myrow & ~15;           // col0 value of the self tile

    for (int col0 = 0; col0 < NPTS; col0 += 16) {
        // B operand: 4x16 f32, column N=lane%16; lanes 0-15 -> (K=0,K=1)=(x,y),
        // lanes 16-31 -> (K=2,K=3)=(z,0). Fully branchless: both loads are
        // unconditionally live (mask multiply), no exec-mask dance.
        int c3 = (col0 + l16) * 3;
        v2f bv;
        bv[0] = P[c3 + kofs];
        bv[1] = P[c3 + 1] * ymask;
        float sqc = SQ[col0 + l16];

        v8f zero = {};
        v8f dot_lo = __builtin_amdgcn_wmma_f32_16x16x4_f32(
            false, a_lo, false, bv, (short)0, zero, false, false);
        v8f dot_hi = __builtin_amdgcn_wmma_f32_16x16x4_f32(
            false, a_hi, false, bv, (short)0, zero, false, false);

        // C/D layout: VGPR r holds M=r (lanes 0-15) / M=r+8 (lanes 16-31),
        // N=lane%16. Store (‖q‖² − 2·p·q); add ‖p‖² per-row during the scan.
        int mbase = hi ? 8 : 0;
#pragma unroll
        for (int r = 0; r < 8; ++r) {
            T[(mbase + r) * LDS_STRIDE + l16]      = sqc - 2.0f * dot_lo[r];
            T[(mbase + r + 16) * LDS_STRIDE + l16] = sqc - 2.0f * dot_hi[r];
        }
        __builtin_amdgcn_wave_barrier();

        // Self-exclusion: exactly one tile per lane contains the self column.
        // Poison it with +INF so the hot scan needs no per-candidate compare.
        if (col0 == diagT) Trow[diagCol] = 3.4e38f;
        __builtin_amdgcn_wave_barrier();

        // Batch-read lane's 16-candidate row into registers (independent DS
        // loads -> one dscnt wait), then scan purely out of VGPRs.
        float cand[16];
#pragma unroll
        for (int cc = 0; cc < 16; ++cc) cand[cc] = Trow[cc];

#pragma unroll
        for (int cc = 0; cc < 16; ++cc) {
            float d = sqr + cand[cc];
            if (d < bd[KNN - 1]) {
                float nd = d;
                int   ni = col0 + cc;
#pragma unroll
                for (int j = 0; j < KNN; ++j) {
                    bool  sw = nd < bd[j];
                    float td = sw ? bd[j] : nd;
                    int   ti = sw ? bi[j] : ni;
                    bd[j] = sw ? nd : bd[j];
                    bi[j] = sw ? ni : bi[j];
                    nd = td; ni = ti;
                }
            }
        }
        __builtin_amdgcn_wave_barrier();
    }

    int* out = knn + ((size_t)b * NPTS + myrow) * KNN;
#pragma unroll
    for (int j = 0; j < KNN; ++j) out[j] = bi[j];
}

// ---------------------------------------------------------------------------
// Kernel 3: uniform Laplacian of both clouds with shared connectivity + L1 sum
// ---------------------------------------------------------------------------
__global__ void PLL_lap_loss_kernel(const float* __restrict__ p1,
                                    const float* __restrict__ p2,
                                    const int* __restrict__ knn,
                                    float* __restrict__ acc) {
    int t = blockIdx.x * blockDim.x + threadIdx.x;
    float v = 0.0f;
    if (t < BATCH * NPTS) {
        int b = t >> 13;            // N = 8192
        int n = t & (NPTS - 1);
        const float* P1 = p1 + (size_t)b * NPTS * 3;
        const float* P2 = p2 + (size_t)b * NPTS * 3;
        const int*   nb = knn + (size_t)t * KNN;
        float s1x = 0, s1y = 0, s1z = 0, s2x = 0, s2y = 0, s2z = 0;
#pragma unroll
        for (int j = 0; j < KNN; ++j) {
            int m = nb[j];
            s1x += P1[m * 3 + 0]; s1y += P1[m * 3 + 1]; s1z += P1[m * 3 + 2];
            s2x += P2[m * 3 + 0]; s2y += P2[m * 3 + 1]; s2z += P2[m * 3 + 2];
        }
        const float inv = 1.0f / (float)KNN;
        float l1x = s1x * inv - P1[n * 3 + 0];
        float l1y = s1y * inv - P1[n * 3 + 1];
        float l1z = s1z * inv - P1[n * 3 + 2];
        float l2x = s2x * inv - P2[n * 3 + 0];
        float l2y = s2y * inv - P2[n * 3 + 1];
        float l2z = s2z * inv - P2[n * 3 + 2];
        v = fabsf(l1x - l2x) + fabsf(l1y - l2y) + fabsf(l1z - l2z);
    }
    // wave32 reduction, then one atomic per wave
#pragma unroll
    for (int o = 16; o > 0; o >>= 1) v += __shfl_down(v, o, 32);
    if ((threadIdx.x & 31) == 0) atomicAdd(acc, v);
}

__global__ void PLL_finalize_kernel(const float* __restrict__ acc, float* __restrict__ out) {
    if (threadIdx.x == 0 && blockIdx.x == 0)
        out[0] = acc[0] * (1.0f / (float)(BATCH * NPTS * 3));
}

// ---------------------------------------------------------------------------
extern "C" void kernel_launch(void* const* d_in, const int* in_sizes, int n_in,
                              void* d_out, int out_size, void* d_ws, size_t ws_size,
                              hipStream_t stream) {
    const float* p1 = (const float*)d_in[0];
    const float* p2 = (const float*)d_in[1];
    float* out = (float*)d_out;

    char*  ws  = (char*)d_ws;
    float* acc = (float*)ws;                                   // 1 float (padded to 256B)
    float* sqn = (float*)(ws + 256);                           // B*N floats   (128 KB)
    int*   knn = (int*)(ws + 256 + (size_t)BATCH * NPTS * 4);  // B*N*K ints   (1 MB)

    hipMemsetAsync(acc, 0, sizeof(float), stream);

    PLL_sqn_kernel<<<(BATCH * NPTS + 255) / 256, 256, 0, stream>>>(p1, sqn);

    PLL_knn_kernel<<<(BATCH * NPTS / 32) / WAVES_PER_BLOCK,
                     WAVES_PER_BLOCK * 32, 0, stream>>>(p1, sqn, knn);

    PLL_lap_loss_kernel<<<(BATCH * NPTS + 255) / 256, 256, 0, stream>>>(p1, p2, knn, acc);

    PLL_finalize_kernel<<<1, 32, 0, stream>>>(acc, out);
}